// MechanismODE_16604343567214
// MI455X (gfx1250) — compile-verified
//
#include <hip/hip_runtime.h>
#include <stdint.h>

// MechanismODE RHS for MI455X (gfx1250, wave32).
//
// out[b,s] = sum_t v_net[b,t] * effects[t,s]   (M=B, K=12, N=16 GEMM)
// Contraction via V_WMMA_F32_16X16X4_F32, K=12 as 3 chained WMMAs.
// One wave = one 16-batch output tile; lanes 0-15 carry K={0,1}, lanes
// 16-31 carry K={2,3} of each K-chunk (ISA 7.12.2 f32 A layout), so each
// lane computes exactly the 6 v_net values its A-fragment needs.
//
// Memory-bound: ~224 B/batch -> ~58.7 MB total -> ~2.5 us at 23.3 TB/s.
// All per-transition operands (rates, gather descriptors, y values) are
// fetched from LDS by runtime index -> pure ds_load addressing, no
// register-array dynamic indexing, no select trees.

typedef __attribute__((ext_vector_type(2))) float v2f;
typedef __attribute__((ext_vector_type(8))) float v8f;

#define S_DIM 16
#define T_DIM 12
#define ROWF 20            // padded y-row stride (80 B: 16B-aligned, conflict-free)
#define ONE_SLOT 16        // yrow[16] == 1.0f -> masked/invalid gathers read 1.0
#define BATCH_PER_BLOCK 128
#define THREADS 256

// ---------------------------------------------------------------------------
// Setup kernel: per-transition descriptor (1 block x 16 threads, cost = noise)
//   desc[t].x = sub | prod<<8 | from<<16 | to<<24   (each in 0..16)
//   desc[t].y = bits(reversible[t] ? 1.0f : 0.0f)
// ---------------------------------------------------------------------------
__global__ void mech_ode_setup(
    const float* __restrict__ stoich,             // [12,16]
    const int* __restrict__ from_idx,             // [12]
    const unsigned char* __restrict__ from_valid, // [12] (jax bool = 1 byte)
    const int* __restrict__ to_idx,               // [12]
    const unsigned char* __restrict__ to_valid,   // [12]
    const unsigned char* __restrict__ reversible, // [12]
    uint2* __restrict__ desc_out)                 // [12] in d_ws
{
    const int t = threadIdx.x;
    if (t >= T_DIM) return;

    // stoich entries are in {-1,0,+1}: at most one substrate, one product.
    int sub = ONE_SLOT, prod = ONE_SLOT;
    for (int s = 0; s < S_DIM; ++s) {
        float w = stoich[t * S_DIM + s];
        if (w < 0.0f) sub  = s;
        if (w > 0.0f) prod = s;
    }
    uint32_t fo = from_valid[t] ? (uint32_t)from_idx[t] : (uint32_t)ONE_SLOT;
    uint32_t to = to_valid[t]   ? (uint32_t)to_idx[t]   : (uint32_t)ONE_SLOT;

    uint2 d;
    d.x = (uint32_t)sub | ((uint32_t)prod << 8) | (fo << 16) | (to << 24);
    d.y = __float_as_uint(reversible[t] ? 1.0f : 0.0f);
    desc_out[t] = d;
}

// ---------------------------------------------------------------------------
// Main kernel
// ---------------------------------------------------------------------------
__global__ __launch_bounds__(THREADS) void mech_ode_wmma(
    const float* __restrict__ y,          // [B,16]
    const float* __restrict__ fwd,        // [B,12]
    const float* __restrict__ rev,        // [B,12]
    const float* __restrict__ effects,    // [12,16]
    const uint2* __restrict__ desc,       // [12] descriptor table (d_ws)
    float* __restrict__ out)              // [B,16]
{
    __shared__ float ytile[BATCH_PER_BLOCK * ROWF];    // 10240 B, padded rows
    __shared__ float ftile[BATCH_PER_BLOCK * T_DIM];   //  6144 B, flat copy
    __shared__ float rtile[BATCH_PER_BLOCK * T_DIM];   //  6144 B, flat copy
    __shared__ uint2 ldsdesc[T_DIM];                   //    96 B

    const int tid = threadIdx.x;
    const int blockBase = blockIdx.x * BATCH_PER_BLOCK;

    // ---- Stage y: 512 coalesced float4 -> padded rows (16B-aligned) ----
    {
        const float4* yg = (const float4*)(y + (size_t)blockBase * S_DIM);
#pragma unroll
        for (int i = 0; i < 2; ++i) {
            const int id  = tid + i * THREADS;          // 0..511
            float4 v      = yg[id];
            *(float4*)&ytile[(id >> 2) * ROWF + (id & 3) * 4] = v;
        }
    }
    // ---- Stage fwd/rev: flat 1536-float slabs (384 float4 each) ----
    {
        const float4* fg = (const float4*)(fwd + (size_t)blockBase * T_DIM);
        const float4* rg = (const float4*)(rev + (size_t)blockBase * T_DIM);
        ((float4*)ftile)[tid] = fg[tid];
        ((float4*)rtile)[tid] = rg[tid];
        if (tid < 128) {
            ((float4*)ftile)[tid + THREADS] = fg[tid + THREADS];
            ((float4*)rtile)[tid + THREADS] = rg[tid + THREADS];
        }
    }
    if (tid < BATCH_PER_BLOCK) ytile[tid * ROWF + ONE_SLOT] = 1.0f;
    if (tid < T_DIM)           ldsdesc[tid] = desc[tid];
    __syncthreads();

    const int wave   = tid >> 5;
    const int lane   = tid & 31;
    const int laneLo = lane & 15;
    const int half   = lane >> 4;          // 0: K {0,1} per chunk, 1: K {2,3}
    const int t0     = 2 * half;

    const int waveBase = blockBase + wave * 16;
    const int row      = wave * 16 + laneLo;      // local batch row

    const float* yrow = &ytile[row * ROWF];
    const float* frow = &ftile[row * T_DIM];
    const float* rrow = &rtile[row * T_DIM];

    // ---- Compute the 6 needed v_net values and feed 3 chained WMMAs ----
    v8f acc = {};
#pragma unroll
    for (int c = 0; c < 3; ++c) {
        float a[2];
#pragma unroll
        for (int j = 0; j < 2; ++j) {
            const int t = 4 * c + t0 + j;             // runtime (half-dependent)
            const uint2 d = ldsdesc[t];               // ds_load_b64
            const float fv = frow[t];                 // ds_load
            const float rv = rrow[t];                 // ds_load
            const uint32_t o = d.x;
            const float rmul = __uint_as_float(d.y);
            const float ysub  = yrow[o & 31];         // 4x ds_load gathers
            const float yprod = yrow[(o >> 8)  & 31];
            const float yf    = yrow[(o >> 16) & 31];
            const float yt    = yrow[(o >> 24) & 31];
            a[j] = fv * yf * ysub - rmul * rv * yt * yprod;
        }
        // B-fragment: rows kb=4c+2*half, +1 at column laneLo (mirrors A layout)
        const int kb = 4 * c + t0;
        float b0 = effects[(kb + 0) * S_DIM + laneLo];
        float b1 = effects[(kb + 1) * S_DIM + laneLo];

        v2f A;  A[0]  = a[0]; A[1]  = a[1];
        v2f Bm; Bm[0] = b0;   Bm[1] = b1;
        acc = __builtin_amdgcn_wmma_f32_16x16x4_f32(
            false, A, false, Bm, (short)0, acc, false, false);
    }

    // ---- Store D: VGPR m -> row waveBase + m + 8*half, col = laneLo ----
    float* orow = out + (size_t)(waveBase + 8 * half) * S_DIM + laneLo;
#pragma unroll
    for (int m = 0; m < 8; ++m) {
        orow[(size_t)m * S_DIM] = acc[m];
    }
}

extern "C" void kernel_launch(void* const* d_in, const int* in_sizes, int n_in,
                              void* d_out, int out_size, void* d_ws, size_t ws_size,
                              hipStream_t stream) {
    // setup_inputs order: t, y, forward_rates, reverse_rates, stoich, effects,
    //                     from_idx, from_valid, to_idx, to_valid, reversible
    const float* y        = (const float*)d_in[1];
    const float* fwd      = (const float*)d_in[2];
    const float* rev      = (const float*)d_in[3];
    const float* stoich   = (const float*)d_in[4];
    const float* effects  = (const float*)d_in[5];
    const int*   from_idx = (const int*)d_in[6];
    const unsigned char* from_valid = (const unsigned char*)d_in[7];
    const int*   to_idx   = (const int*)d_in[8];
    const unsigned char* to_valid   = (const unsigned char*)d_in[9];
    const unsigned char* reversible = (const unsigned char*)d_in[10];
    float* out = (float*)d_out;

    uint2* desc = (uint2*)d_ws;   // 12 x 8 B descriptor table

    mech_ode_setup<<<1, 16, 0, stream>>>(
        stoich, from_idx, from_valid, to_idx, to_valid, reversible, desc);

    const int B = in_sizes[1] / S_DIM;              // 262144
    const int grid = B / BATCH_PER_BLOCK;           // B divisible by 128

    mech_ode_wmma<<<grid, THREADS, 0, stream>>>(
        y, fwd, rev, effects, desc, out);
}